// DecodePredictions_12867722019202
// MI455X (gfx1250) — compile-verified
//
#include <hip/hip_runtime.h>
#include <hip/hip_bf16.h>
#include <math.h>

// ---------------- problem constants (from reference) ----------------
constexpr int    NA      = 1000000;   // N_ANCHORS
constexpr int    NC      = 8;         // NUM_CLASSES
constexpr float  CONF_T  = 0.05f;
constexpr float  IOU_T   = 0.35f;
constexpr float  NEGC    = -1e30f;
constexpr float  EPS_T   = 1e-8f;

// ---------------- selection parameters ----------------
constexpr int    KSEL    = 1536;      // target top-K per class (>= 100 picks + suppressed)
constexpr int    MCAP    = 4096;      // candidate capacity per class (>= KSEL + boundary bin)

// ---------------- workspace layout (bytes) ----------------
constexpr size_t OFF_SCORES = 0;                                      // 8*NA f32 masked scores
constexpr size_t OFF_HIST   = OFF_SCORES + (size_t)NC * NA * 4;       // 8*65536 u32 histogram
constexpr size_t OFF_CNT    = OFF_HIST   + (size_t)NC * 65536 * 4;    // 8 u32 candidate counters
constexpr size_t OFF_CUT    = OFF_CNT    + (size_t)NC * 4;            // 8 u32 key cutoffs
constexpr size_t OFF_CIDX   = OFF_CUT    + (size_t)NC * 4;            // 8*MCAP u32 candidate indices
constexpr size_t OFF_SW     = OFF_CIDX   + (size_t)NC * MCAP * 4;     // 800 f32 per-class scores (16B aligned)
constexpr size_t OFF_BW     = OFF_SW     + (size_t)NC * 100 * 4;      // 800*6 f32 per-class boxes

// ---------------- CDNA5 async-to-LDS (guarded) ----------------
#if defined(__has_builtin)
#  if __has_builtin(__builtin_amdgcn_global_load_async_to_lds_b32)
#    define HAS_ASYNC_LDS 1
#  endif
#  if __has_builtin(__builtin_amdgcn_global_load_async_to_lds_b128)
#    define HAS_ASYNC_LDS128 1
#  endif
#endif

typedef int v4i __attribute__((vector_size(16)));
typedef __attribute__((address_space(1))) int as1_int;   // global ("__device__")
typedef __attribute__((address_space(3))) int as3_int;   // LDS ("__shared__")
typedef __attribute__((address_space(1))) v4i as1_v4i;   // global 16B vector
typedef __attribute__((address_space(3))) v4i as3_v4i;   // LDS 16B vector

__device__ __forceinline__ void gather_score_to_lds(const float* g, float* l) {
#ifdef HAS_ASYNC_LDS
  __builtin_amdgcn_global_load_async_to_lds_b32(
      (as1_int*)(void*)const_cast<float*>(g),
      (as3_int*)(void*)l,
      0, 0);
#else
  *l = *g;
#endif
}

__device__ __forceinline__ void copy16_to_lds(const float* g, float* l) {
#ifdef HAS_ASYNC_LDS128
  __builtin_amdgcn_global_load_async_to_lds_b128(
      (as1_v4i*)(void*)const_cast<float*>(g),
      (as3_v4i*)(void*)l,
      0, 0);
#else
  l[0] = g[0]; l[1] = g[1]; l[2] = g[2]; l[3] = g[3];
#endif
}

__device__ __forceinline__ void async_wait_all() {
#if defined(HAS_ASYNC_LDS) || defined(HAS_ASYNC_LDS128)
  asm volatile("s_wait_asynccnt 0" ::: "memory");
#endif
}

// order-preserving float -> u32 key (monotone increasing)
__device__ __forceinline__ unsigned fkey(float f) {
  unsigned u = __float_as_uint(f);
  return (u & 0x80000000u) ? ~u : (u | 0x80000000u);
}

// block max over packed u64 keys; requires blockDim.x == 1024 (32 wave32s)
__device__ __forceinline__ unsigned long long block_max_u64(
    unsigned long long v, unsigned long long* swr, unsigned long long* sbest,
    int lane, int wid) {
#pragma unroll
  for (int off = 16; off > 0; off >>= 1) {
    unsigned long long o = __shfl_xor(v, off, 32);
    v = (o > v) ? o : v;
  }
  if (lane == 0) swr[wid] = v;
  __syncthreads();
  if (wid == 0) {
    unsigned long long w = swr[lane];
#pragma unroll
    for (int off = 16; off > 0; off >>= 1) {
      unsigned long long o = __shfl_xor(w, off, 32);
      w = (o > w) ? o : w;
    }
    if (lane == 0) *sbest = w;
  }
  __syncthreads();
  return *sbest;
}

// ---------------------------------------------------------------------------
// Kernel 1: decode + sigmoid + mask + per-class 16-bit-key histogram
// ---------------------------------------------------------------------------
__global__ void k_decode_hist(const float* __restrict__ pred,
                              const float* __restrict__ anc,
                              const float* __restrict__ var,
                              float* __restrict__ scores8,
                              unsigned* __restrict__ hist) {
  int i = blockIdx.x * blockDim.x + threadIdx.x;
  if (i >= NA) return;
  const float v0 = var[0], v1 = var[1], v2 = var[2];
  const float v3 = var[3], v4 = var[4], v5 = var[5];
  const float* a = anc + (size_t)i * 6;
  const float* p = pred + (size_t)i * 14;
  float asx = a[3], asy = a[4], asz = a[5];
  float cx = p[8]  * v0 * asx + a[0];
  float cy = p[9]  * v1 * asy + a[1];
  float cz = p[10] * v2 * asz + a[2];
  float dx = expf(p[11] * v3) * asx;
  float dy = expf(p[12] * v4) * asy;
  float dz = expf(p[13] * v5) * asz;
  bool pos = (cx > 0.f) && (cy > 0.f) && (cz > 0.f) &&
             (dx > 0.f) && (dy > 0.f) && (dz > 0.f);
#pragma unroll
  for (int c = 0; c < NC; ++c) {
    float s = 1.0f / (1.0f + expf(-p[c]));
    bool valid = (s > CONF_T) && pos;
    float out = valid ? s : NEGC;
    scores8[(size_t)c * NA + i] = out;
    if (valid) atomicAdd(hist + ((size_t)c << 16) + (fkey(s) >> 16), 1u);
  }
}

// ---------------------------------------------------------------------------
// Kernel 2: per-class cutoff = largest 16-bit key bin with suffix count >= KSEL
// ---------------------------------------------------------------------------
__global__ void k_cutoff(const unsigned* __restrict__ hist,
                         unsigned* __restrict__ cutoff) {
  __shared__ unsigned part[256];
  const int c = blockIdx.x;
  const unsigned* h = hist + (size_t)c * 65536;
  unsigned sum = 0;
  const int base = threadIdx.x * 256;
  for (int b = 0; b < 256; ++b) sum += h[base + b];
  part[threadIdx.x] = sum;
  __syncthreads();
  if (threadIdx.x == 0) {
    unsigned acc = 0, cut = 0;
    int j;
    for (j = 255; j >= 0; --j) {
      if (acc + part[j] >= (unsigned)KSEL) break;
      acc += part[j];
    }
    if (j >= 0) {
      for (int b = j * 256 + 255; b >= j * 256; --b) {
        acc += h[b];
        if (acc >= (unsigned)KSEL) { cut = (unsigned)b; break; }
      }
    }
    cutoff[c] = cut;   // cut==0 => keep all valid
  }
}

// ---------------------------------------------------------------------------
// Kernel 3: compact candidate indices above cutoff
// ---------------------------------------------------------------------------
__global__ void k_compact(const float* __restrict__ scores8,
                          const unsigned* __restrict__ cutoff,
                          unsigned* __restrict__ cnt,
                          unsigned* __restrict__ candidx) {
  const int stride = gridDim.x * blockDim.x;
  for (int i = blockIdx.x * blockDim.x + threadIdx.x; i < NA; i += stride) {
#pragma unroll
    for (int c = 0; c < NC; ++c) {
      const float* sp = scores8 + (size_t)c * NA + i;
      __builtin_prefetch(sp + stride, 0, 0);   // -> global_prefetch_b8
      float s = *sp;
      if (s > CONF_T && (fkey(s) >> 16) >= cutoff[c]) {
        unsigned pos = atomicAdd(cnt + c, 1u);
        if (pos < (unsigned)MCAP) candidx[(size_t)c * MCAP + pos] = (unsigned)i;
      }
    }
  }
}

// ---------------------------------------------------------------------------
// Kernel 4: per-class greedy NMS, fully LDS-resident (CDNA5: 320KB/WG LDS)
//   scores (16KB) async-gathered into LDS, boxes (112KB) decoded into LDS
// ---------------------------------------------------------------------------
__global__ __launch_bounds__(1024) void k_nms(
    const float* __restrict__ pred, const float* __restrict__ anc,
    const float* __restrict__ var, const float* __restrict__ scores8,
    const unsigned* __restrict__ cnt, const unsigned* __restrict__ candidx,
    float* __restrict__ Sw, float* __restrict__ Bw) {
  __shared__ float ssc[MCAP];            // 16 KB
  __shared__ float cbx[MCAP * 7];        // 112 KB: inf3, sup3, vol (stride 7: bank-conflict-free)
  __shared__ unsigned long long swr[32];
  __shared__ unsigned long long sbest;
  __shared__ float swin[7];

  const int c = blockIdx.x;
  const int tid = threadIdx.x, lane = tid & 31, wid = tid >> 5;
  const int n = min((int)cnt[c], MCAP);

  const float v0 = var[0], v1 = var[1], v2 = var[2];
  const float v3 = var[3], v4 = var[4], v5 = var[5];

  // init unused slots (no overlap with async destinations)
  for (int s = tid; s < MCAP; s += 1024)
    if (s >= n) ssc[s] = NEGC;

  // gather candidate scores into LDS + decode candidate boxes into LDS
  for (int s = tid; s < n; s += 1024) {
    unsigned idx = candidx[(size_t)c * MCAP + s];
    gather_score_to_lds(scores8 + (size_t)c * NA + idx, &ssc[s]);
    const float* a = anc + (size_t)idx * 6;
    const float* p = pred + (size_t)idx * 14;
    float asx = a[3], asy = a[4], asz = a[5];
    float cx = p[8]  * v0 * asx + a[0];
    float cy = p[9]  * v1 * asy + a[1];
    float cz = p[10] * v2 * asz + a[2];
    float dx = expf(p[11] * v3) * asx;
    float dy = expf(p[12] * v4) * asy;
    float dz = expf(p[13] * v5) * asz;
    float* cb = cbx + (size_t)s * 7;
    cb[0] = cx - 0.5f * dx; cb[1] = cy - 0.5f * dy; cb[2] = cz - 0.5f * dz;
    cb[3] = cx + 0.5f * dx; cb[4] = cy + 0.5f * dy; cb[5] = cz + 0.5f * dz;
    cb[6] = dx * dy * dz;
  }
  async_wait_all();
  __syncthreads();

  for (int t = 0; t < 100; ++t) {
    // ---- argmax over LDS scores ----
    unsigned long long lm = ((unsigned long long)fkey(NEGC) << 32);
    for (int s = tid; s < n; s += 1024) {
      unsigned long long k =
          ((unsigned long long)fkey(ssc[s]) << 32) | (unsigned)s;
      if (k > lm) lm = k;
    }
    unsigned long long best = block_max_u64(lm, swr, &sbest, lane, wid);
    int bslot = (int)(unsigned)(best & 0xffffffffull);
    float bsc = ssc[bslot];
    bool valid = bsc > (NEGC * 0.5f);

    if (tid == 0) {
      const float* wb = cbx + (size_t)bslot * 7;
      float i0 = wb[0], i1 = wb[1], i2 = wb[2];
      float s0 = wb[3], s1 = wb[4], s2 = wb[5], vv = wb[6];
      swin[0] = i0; swin[1] = i1; swin[2] = i2;
      swin[3] = s0; swin[4] = s1; swin[5] = s2; swin[6] = vv;
      Sw[c * 100 + t] = valid ? bsc : NEGC;
      float* B = Bw + (size_t)(c * 100 + t) * 6;
      B[0] = valid ? (i0 + s0) * 0.5f : 0.f;
      B[1] = valid ? (i1 + s1) * 0.5f : 0.f;
      B[2] = valid ? (i2 + s2) * 0.5f : 0.f;
      B[3] = valid ? (s0 - i0) : 0.f;
      B[4] = valid ? (s1 - i1) : 0.f;
      B[5] = valid ? (s2 - i2) : 0.f;
    }
    __syncthreads();

    // ---- suppression (includes self: IoU == 1) ----
    if (valid) {
      float ai0 = swin[0], ai1 = swin[1], ai2 = swin[2];
      float as0 = swin[3], as1 = swin[4], as2 = swin[5], av = swin[6];
      for (int s = tid; s < n; s += 1024) {
        if (ssc[s] > (NEGC * 0.5f)) {
          const float* bb = cbx + (size_t)s * 7;
          float ex = fmaxf(fminf(as0, bb[3]) - fmaxf(ai0, bb[0]), 0.f);
          float ey = fmaxf(fminf(as1, bb[4]) - fmaxf(ai1, bb[1]), 0.f);
          float ez = fmaxf(fminf(as2, bb[5]) - fmaxf(ai2, bb[2]), 0.f);
          float inter = ex * ey * ez;
          float uni = fmaxf(av + bb[6] - inter, EPS_T);
          if (inter >= IOU_T * uni) ssc[s] = NEGC;
        }
      }
    }
    __syncthreads();
  }
}

// ---------------------------------------------------------------------------
// Kernel 5: global top-100 of the 800 per-class detections
// ---------------------------------------------------------------------------
__global__ __launch_bounds__(1024) void k_finalize(
    const float* __restrict__ Sw, const float* __restrict__ Bw,
    float* __restrict__ out) {
  __shared__ alignas(16) float ls[800];
  __shared__ unsigned long long swr[32];
  __shared__ unsigned long long sbest;
  const int tid = threadIdx.x, lane = tid & 31, wid = tid >> 5;
  // bulk async stage: 200 lanes x 16B (Sw is 16B-aligned in workspace)
  if (tid < 200) copy16_to_lds(Sw + tid * 4, &ls[tid * 4]);
  async_wait_all();
  __syncthreads();
  for (int t = 0; t < 100; ++t) {
    unsigned long long k = 0;
    if (tid < 800)
      k = ((unsigned long long)fkey(ls[tid]) << 32) | (unsigned)(1023 - tid);
    unsigned long long best = block_max_u64(k, swr, &sbest, lane, wid);
    int j = 1023 - (int)(unsigned)(best & 0xffffffffull);
    float sc = ls[j];
    bool ok = sc > (NEGC * 0.5f);
    if (tid == 0) {
      out[600 + t] = ok ? sc : 0.f;
      out[700 + t] = ok ? (float)(j / 100) : 0.f;
      const float* B = Bw + (size_t)j * 6;
#pragma unroll
      for (int d = 0; d < 6; ++d) out[t * 6 + d] = ok ? B[d] : 0.f;
      ls[j] = NEGC;
    }
    __syncthreads();
  }
}

// ---------------------------------------------------------------------------
extern "C" void kernel_launch(void* const* d_in, const int* in_sizes, int n_in,
                              void* d_out, int out_size, void* d_ws,
                              size_t ws_size, hipStream_t stream) {
  (void)in_sizes; (void)n_in; (void)out_size; (void)ws_size;
  const float* pred = (const float*)d_in[0];
  const float* anc  = (const float*)d_in[1];
  const float* var  = (const float*)d_in[2];
  char* ws = (char*)d_ws;
  float*    scores8 = (float*)(ws + OFF_SCORES);
  unsigned* hist    = (unsigned*)(ws + OFF_HIST);
  unsigned* cnt     = (unsigned*)(ws + OFF_CNT);
  unsigned* cut     = (unsigned*)(ws + OFF_CUT);
  unsigned* cidx    = (unsigned*)(ws + OFF_CIDX);
  float*    Sw      = (float*)(ws + OFF_SW);
  float*    Bw      = (float*)(ws + OFF_BW);
  float*    out     = (float*)d_out;

  // zero histogram + counters (idempotent per launch; graph-capturable)
  (void)hipMemsetAsync(hist, 0, (size_t)NC * 65536 * 4 + NC * 4, stream);

  k_decode_hist<<<(NA + 255) / 256, 256, 0, stream>>>(pred, anc, var, scores8, hist);
  k_cutoff<<<NC, 256, 0, stream>>>(hist, cut);
  k_compact<<<1024, 256, 0, stream>>>(scores8, cut, cnt, cidx);
  k_nms<<<NC, 1024, 0, stream>>>(pred, anc, var, scores8, cnt, cidx, Sw, Bw);
  k_finalize<<<1, 1024, 0, stream>>>(Sw, Bw, out);
}